// BahdanauAttention_60902636257371
// MI455X (gfx1250) — compile-verified
//
#include <hip/hip_runtime.h>
#include <math.h>

typedef __attribute__((ext_vector_type(2))) float v2f;
typedef __attribute__((ext_vector_type(8))) float v8f;
typedef __attribute__((ext_vector_type(4))) int   v4i;

#define BB   128
#define TT   1024
#define ED   512      // ENC_D
#define AD   512      // ATT_D
#define DD   512      // DEC_D
#define ENC_LD 520    // padded LDS row stride (520 % 64 == 8 -> no bank conflicts)
#define MROWS 64      // t-rows per block (4 t-tiles of 16)
#define ENERGY_SMEM ((MROWS * ENC_LD + MROWS) * sizeof(float))

// ---- CDNA5 hardware tanh (V_TANH_F32) if the toolchain exposes it ---------
#if __has_builtin(__builtin_amdgcn_tanhf)
static __device__ __forceinline__ float fast_tanh(float x) { return __builtin_amdgcn_tanhf(x); }
#elif __has_builtin(__builtin_amdgcn_tanh_f32)
static __device__ __forceinline__ float fast_tanh(float x) { return __builtin_amdgcn_tanh_f32(x); }
#else
static __device__ __forceinline__ float fast_tanh(float x) { return tanhf(x); }
#endif

// ---- CDNA5 async global->LDS copy (ASYNCcnt-tracked) if available ---------
#if __has_builtin(__builtin_amdgcn_global_load_async_to_lds_b128) && \
    __has_builtin(__builtin_amdgcn_s_wait_asynccnt)
#define HAVE_ASYNC_LDS 1
typedef __attribute__((address_space(1))) v4i* gv4i_p;
typedef __attribute__((address_space(3))) v4i* lv4i_p;
#else
#define HAVE_ASYNC_LDS 0
#endif

// ---------------------------------------------------------------------------
// Kernel 1: dec_proj[b, a] = decoder_hidden[b, :] . W_dec[a, :]
// ---------------------------------------------------------------------------
__global__ __launch_bounds__(256) void decproj_kernel(
    const float* __restrict__ dec_h, const float* __restrict__ W_dec,
    float* __restrict__ dec_proj)
{
    const int b = blockIdx.x;
    __shared__ __align__(16) float sh[DD];
    for (int i = threadIdx.x; i < DD; i += 256) sh[i] = dec_h[(size_t)b * DD + i];
    __syncthreads();
    for (int a = threadIdx.x; a < AD; a += 256) {
        const float* w = W_dec + (size_t)a * DD;
        float acc = 0.f;
        #pragma unroll 4
        for (int k = 0; k < DD; k += 4) {
            float4 wv = *(const float4*)(w + k);
            acc = fmaf(wv.x, sh[k + 0], acc);
            acc = fmaf(wv.y, sh[k + 1], acc);
            acc = fmaf(wv.z, sh[k + 2], acc);
            acc = fmaf(wv.w, sh[k + 3], acc);
        }
        dec_proj[(size_t)b * AD + a] = acc;
    }
}

// ---------------------------------------------------------------------------
// Kernel 2 (WMMA core):
//   energy[b, t] = sum_a v[a] * tanh( enc_proj[b,t,a] + dec_proj[b,a] )
// One block = one (b, 64-row t-block). 8 waves; each wave owns 4 a-tiles of 16
// and 4 t-tiles of 16 -> 16 concurrent 16x16 accumulators. Each B fragment
// (W_enc, L2-resident) feeds 4 WMMAs; each A fragment (LDS) feeds 4 WMMAs.
// W_enc L2 traffic per block amortized over 64 rows instead of 16.
// ---------------------------------------------------------------------------
__global__ __launch_bounds__(256) void energy_kernel(
    const float* __restrict__ enc, const float* __restrict__ W_enc,
    const float* __restrict__ v,   const float* __restrict__ dec_proj,
    float* __restrict__ energy)
{
    const int blk = blockIdx.x;
    const int b   = blk >> 4;             // TT/MROWS = 16 t-blocks per batch row
    const int t0  = (blk & 15) << 6;

    extern __shared__ float smem[];
    float* encT    = smem;                 // [MROWS * ENC_LD]
    float* sEnergy = smem + MROWS * ENC_LD; // [MROWS]

    const int tid = threadIdx.x;
    if (tid < MROWS) sEnergy[tid] = 0.f;

    // Stage the 64 x 512 encoder tile into LDS (32 x b128 per thread).
    const float* encPtr = enc + ((size_t)b * TT + t0) * ED;
#if HAVE_ASYNC_LDS
    for (int i = tid; i < MROWS * (ED / 4); i += 256) {
        const int row = i >> 7;            // 128 float4 per row
        const int c4  = (i & 127) << 2;
        __builtin_amdgcn_global_load_async_to_lds_b128(
            (gv4i_p)(encPtr + row * ED + c4),
            (lv4i_p)(&encT[row * ENC_LD + c4]),
            0, 0);
    }
    __builtin_amdgcn_s_wait_asynccnt(0);   // my wave's async copies done
    __syncthreads();                        // all waves' copies published
#else
    for (int i = tid; i < MROWS * (ED / 4); i += 256) {
        const int row = i >> 7;
        const int c4  = (i & 127) << 2;
        *(float4*)(&encT[row * ENC_LD + c4]) = *(const float4*)(encPtr + row * ED + c4);
    }
    __syncthreads();
#endif

    const int lane = tid & 31;
    const int wave = tid >> 5;
    const int half = lane >> 4;          // 0: K{0,1}/rows 0..7, 1: K{2,3}/rows 8..15
    const int lmod = lane & 15;

    const int aBase = wave << 6;         // this wave covers a in [aBase, aBase+64)
    const float* wrow0 = W_enc + (size_t)(aBase +  0 + lmod) * ED + half * 2;
    const float* wrow1 = W_enc + (size_t)(aBase + 16 + lmod) * ED + half * 2;
    const float* wrow2 = W_enc + (size_t)(aBase + 32 + lmod) * ED + half * 2;
    const float* wrow3 = W_enc + (size_t)(aBase + 48 + lmod) * ED + half * 2;
    const float* arow  = &encT[lmod * ENC_LD + half * 2];

    v8f acc[4][4];                       // [t-tile][a-tile]
    #pragma unroll
    for (int tt = 0; tt < 4; ++tt)
        #pragma unroll
        for (int j = 0; j < 4; ++j) acc[tt][j] = (v8f){};

    #pragma unroll 2
    for (int k = 0; k < ED; k += 4) {
        const v2f B0 = *(const v2f*)(wrow0 + k);
        const v2f B1 = *(const v2f*)(wrow1 + k);
        const v2f B2 = *(const v2f*)(wrow2 + k);
        const v2f B3 = *(const v2f*)(wrow3 + k);
        #pragma unroll
        for (int tt = 0; tt < 4; ++tt) {
            const v2f Af = *(const v2f*)(arow + tt * 16 * ENC_LD + k);
            acc[tt][0] = __builtin_amdgcn_wmma_f32_16x16x4_f32(false, Af, false, B0, (short)0, acc[tt][0], false, false);
            acc[tt][1] = __builtin_amdgcn_wmma_f32_16x16x4_f32(false, Af, false, B1, (short)0, acc[tt][1], false, false);
            acc[tt][2] = __builtin_amdgcn_wmma_f32_16x16x4_f32(false, Af, false, B2, (short)0, acc[tt][2], false, false);
            acc[tt][3] = __builtin_amdgcn_wmma_f32_16x16x4_f32(false, Af, false, B3, (short)0, acc[tt][3], false, false);
        }
    }

    // D layout: lane holds column N = lmod, rows M = half*8 + r (per t-tile).
    const float* dpB = dec_proj + (size_t)b * AD + aBase + lmod;
    const float* vB  = v + aBase + lmod;
    const float dp0 = dpB[0],  dp1 = dpB[16], dp2 = dpB[32], dp3 = dpB[48];
    const float vv0 = vB[0],   vv1 = vB[16],  vv2 = vB[32],  vv3 = vB[48];

    #pragma unroll
    for (int tt = 0; tt < 4; ++tt) {
        float part[8];
        #pragma unroll
        for (int r = 0; r < 8; ++r)
            part[r] = vv0 * fast_tanh(acc[tt][0][r] + dp0)
                    + vv1 * fast_tanh(acc[tt][1][r] + dp1)
                    + vv2 * fast_tanh(acc[tt][2][r] + dp2)
                    + vv3 * fast_tanh(acc[tt][3][r] + dp3);
        // Reduce over the a (column) dimension within each 16-lane half.
        #pragma unroll
        for (int m = 1; m < 16; m <<= 1) {
            #pragma unroll
            for (int r = 0; r < 8; ++r) part[r] += __shfl_xor(part[r], m, 32);
        }
        if (lmod == 0) {
            #pragma unroll
            for (int r = 0; r < 8; ++r)
                atomicAdd(&sEnergy[tt * 16 + half * 8 + r], part[r]);
        }
    }
    __syncthreads();
    if (tid < MROWS) energy[(size_t)b * TT + t0 + tid] = sEnergy[tid];
}

// ---------------------------------------------------------------------------
// Kernel 3: softmax over T per batch row; also zero-fills the context output
// so kernel 4 can accumulate with atomics (graph-capture safe, no memset).
// ---------------------------------------------------------------------------
__global__ __launch_bounds__(256) void softmax_zero_kernel(
    const float* __restrict__ energy, float* __restrict__ attn,
    float* __restrict__ ctx)
{
    const int b   = blockIdx.x;
    const int tid = threadIdx.x;
    const int lane = tid & 31, wave = tid >> 5;
    const float* e = energy + (size_t)b * TT;

    __shared__ float sred[8];
    float vals[4];
    float mx = -3.402823466e38f;
    #pragma unroll
    for (int j = 0; j < 4; ++j) { vals[j] = e[tid + j * 256]; mx = fmaxf(mx, vals[j]); }
    #pragma unroll
    for (int m = 16; m >= 1; m >>= 1) mx = fmaxf(mx, __shfl_xor(mx, m, 32));
    if (lane == 0) sred[wave] = mx;
    __syncthreads();
    float bm = sred[0];
    #pragma unroll
    for (int w = 1; w < 8; ++w) bm = fmaxf(bm, sred[w]);

    float s = 0.f;
    #pragma unroll
    for (int j = 0; j < 4; ++j) { vals[j] = __expf(vals[j] - bm); s += vals[j]; }
    #pragma unroll
    for (int m = 16; m >= 1; m >>= 1) s += __shfl_xor(s, m, 32);
    __syncthreads();
    if (lane == 0) sred[wave] = s;
    __syncthreads();
    float tot = 0.f;
    #pragma unroll
    for (int w = 0; w < 8; ++w) tot += sred[w];
    const float inv = 1.0f / tot;

    #pragma unroll
    for (int j = 0; j < 4; ++j) attn[(size_t)b * TT + tid + j * 256] = vals[j] * inv;

    // zero context accumulation region (exactly covers B*ED with 128x256 threads x2)
    ctx[(size_t)b * ED + tid]       = 0.f;
    ctx[(size_t)b * ED + tid + 256] = 0.f;
}

// ---------------------------------------------------------------------------
// Kernel 4: context[b, e] += sum_t attn[b, t] * enc[b, t, e]
// Grid B*4; each block streams a 256-row t-chunk, coalesced over e.
// ---------------------------------------------------------------------------
__global__ __launch_bounds__(256) void context_kernel(
    const float* __restrict__ enc, const float* __restrict__ attn,
    float* __restrict__ ctx)
{
    const int b  = blockIdx.x >> 2;
    const int t0 = (blockIdx.x & 3) << 8;
    const int tid = threadIdx.x;

    __shared__ float sa[256];
    sa[tid] = attn[(size_t)b * TT + t0 + tid];
    __syncthreads();

    const float* ep = enc + ((size_t)b * TT + t0) * ED;
    float a0 = 0.f, a1 = 0.f;
    #pragma unroll 4
    for (int t = 0; t < 256; ++t) {
        const float w = sa[t];
        a0 = fmaf(w, ep[(size_t)t * ED + tid],       a0);
        a1 = fmaf(w, ep[(size_t)t * ED + tid + 256], a1);
    }
    atomicAdd(&ctx[(size_t)b * ED + tid],       a0);
    atomicAdd(&ctx[(size_t)b * ED + tid + 256], a1);
}

// ---------------------------------------------------------------------------
extern "C" void kernel_launch(void* const* d_in, const int* in_sizes, int n_in,
                              void* d_out, int out_size, void* d_ws, size_t ws_size,
                              hipStream_t stream)
{
    const float* enc   = (const float*)d_in[0];  // [B, T, ED]
    const float* dech  = (const float*)d_in[1];  // [B, DD]
    const float* W_enc = (const float*)d_in[2];  // [AD, ED]
    const float* W_dec = (const float*)d_in[3];  // [AD, DD]
    const float* v     = (const float*)d_in[4];  // [AD]

    float* out  = (float*)d_out;
    float* ctx  = out;                         // [B, ED]
    float* attn = out + (size_t)BB * ED;       // [B, T]

    float* dec_proj = (float*)d_ws;                       // [B, AD]
    float* energy   = dec_proj + (size_t)BB * AD;         // [B, T]

    decproj_kernel     <<<BB,                256, 0, stream>>>(dech, W_dec, dec_proj);
    energy_kernel      <<<BB * (TT/MROWS),   256, ENERGY_SMEM, stream>>>(enc, W_enc, v, dec_proj, energy);
    softmax_zero_kernel<<<BB,                256, 0, stream>>>(energy, attn, ctx);
    context_kernel     <<<BB * 4,            256, 0, stream>>>(enc, attn, ctx);
}